// SoftShapeNet_9251359556181
// MI455X (gfx1250) — compile-verified
//
#include <hip/hip_runtime.h>
#include <hip/hip_bf16.h>

typedef __attribute__((ext_vector_type(2))) float v2f;
typedef __attribute__((ext_vector_type(8))) float v8f;

#define NTOK  8192
#define DDIM  512
#define EEXP  8
#define HDIM  2048
#define MT    32          // tokens per FFN tile
#define TPE   256         // upper-bound tiles per expert = ceil(NTOK/MT)
#define HCS   132         // padded LDS stride for Hc (128 + 4)

// gelu (tanh approx), branch-free: 0.5*v*(1+tanh(u)) = v * sigmoid(2u)
// division replaced by single-instruction v_rcp_f32 (~1 ulp)
__device__ __forceinline__ float gelu_tanh(float v) {
    const float c0 = 1.5957691216057308f;   // 2*sqrt(2/pi)
    float u = c0 * (v + 0.044715f * v * v * v);
    return v * __builtin_amdgcn_rcpf(1.0f + __expf(-u));
}

// ---------------- 1. zero scratch (128-bit stores) ----------------
__global__ void zero_kernel(float4* __restrict__ p, size_t n4) {
    size_t i = (size_t)blockIdx.x * blockDim.x + threadIdx.x;
    size_t stride = (size_t)gridDim.x * blockDim.x;
    float4 z; z.x = 0.f; z.y = 0.f; z.z = 0.f; z.w = 0.f;
    for (; i < n4; i += stride) p[i] = z;
}

// ---------------- 2. gating: one wave per token ----------------
__global__ void gate_kernel(const float* __restrict__ x, const float* __restrict__ wg,
                            int* __restrict__ tok_idx, float* __restrict__ tok_gate) {
    int wavesPerBlock = blockDim.x >> 5;
    int n = blockIdx.x * wavesPerBlock + (threadIdx.x >> 5);
    if (n >= NTOK) return;
    int lane = threadIdx.x & 31;

    float acc[EEXP];
#pragma unroll
    for (int e = 0; e < EEXP; ++e) acc[e] = 0.0f;

    const float* xr = x + (size_t)n * DDIM;
    for (int d = lane; d < DDIM; d += 32) {
        float xv = xr[d];
        const float* wr = wg + (size_t)d * EEXP;
#pragma unroll
        for (int e = 0; e < EEXP; ++e) acc[e] += xv * wr[e];
    }
#pragma unroll
    for (int off = 16; off > 0; off >>= 1) {
#pragma unroll
        for (int e = 0; e < EEXP; ++e) acc[e] += __shfl_xor(acc[e], off, 32);
    }
    if (lane == 0) {
        float m = acc[0];
#pragma unroll
        for (int e = 1; e < EEXP; ++e) m = fmaxf(m, acc[e]);
        float s = 0.0f, p[EEXP];
#pragma unroll
        for (int e = 0; e < EEXP; ++e) { p[e] = __expf(acc[e] - m); s += p[e]; }
        float inv = 1.0f / s;
#pragma unroll
        for (int e = 0; e < EEXP; ++e) p[e] *= inv;

        int i0 = 0; float v0 = p[0];
#pragma unroll
        for (int e = 1; e < EEXP; ++e) if (p[e] > v0) { v0 = p[e]; i0 = e; }
        int i1 = -1; float v1 = -1.0f;
#pragma unroll
        for (int e = 0; e < EEXP; ++e) if (e != i0 && p[e] > v1) { v1 = p[e]; i1 = e; }

        float denom = v0 + v1 + 1e-6f;
        tok_idx[2 * n]     = i0;
        tok_idx[2 * n + 1] = i1;
        tok_gate[2 * n]     = v0 / denom;
        tok_gate[2 * n + 1] = v1 / denom;
    }
}

// ---------------- 3. per-expert stats (deterministic tree reduce) ----------------
__global__ void stats_kernel(const int* __restrict__ tok_idx, const float* __restrict__ tok_gate,
                             int* __restrict__ counts, float* __restrict__ imp,
                             float* __restrict__ loadv) {
    int e = blockIdx.x;
    int tid = threadIdx.x;
    float s = 0.0f; int c = 0;
    for (int i = tid; i < 2 * NTOK; i += blockDim.x) {
        if (tok_idx[i] == e) { s += tok_gate[i]; c += 1; }
    }
    __shared__ float ssum[256];
    __shared__ int   csum[256];
    ssum[tid] = s; csum[tid] = c;
    __syncthreads();
    for (int st = 128; st > 0; st >>= 1) {
        if (tid < st) { ssum[tid] += ssum[tid + st]; csum[tid] += csum[tid + st]; }
        __syncthreads();
    }
    if (tid == 0) {
        counts[e] = csum[0];
        imp[e]    = ssum[0];
        loadv[e]  = (float)csum[0];
    }
}

// ---------------- 4. offsets scan + aux loss ----------------
__device__ __forceinline__ float cv_sq(const float* v) {
    float mean = 0.0f;
#pragma unroll
    for (int e = 0; e < EEXP; ++e) mean += v[e];
    mean *= (1.0f / EEXP);
    float var = 0.0f;
#pragma unroll
    for (int e = 0; e < EEXP; ++e) { float d = v[e] - mean; var += d * d; }
    var *= (1.0f / (EEXP - 1));
    return var / (mean * mean + 1e-10f);
}

__global__ void scan_loss_kernel(const int* __restrict__ counts, int* __restrict__ offsets,
                                 int* __restrict__ cursor, const float* __restrict__ imp,
                                 const float* __restrict__ loadv, float* __restrict__ loss_out) {
    if (threadIdx.x == 0) {
        int off = 0;
        for (int e = 0; e < EEXP; ++e) { offsets[e] = off; off += counts[e]; cursor[e] = 0; }
        float vi[EEXP], vl[EEXP];
        for (int e = 0; e < EEXP; ++e) { vi[e] = imp[e]; vl[e] = loadv[e]; }
        loss_out[0] = cv_sq(vi) + cv_sq(vl);
    }
}

// ---------------- 5. scatter tokens into expert bins ----------------
__global__ void scatter_kernel(const int* __restrict__ tok_idx, const float* __restrict__ tok_gate,
                               const int* __restrict__ offsets, int* __restrict__ cursor,
                               int* __restrict__ a_tok, float* __restrict__ a_gate) {
    int n = blockIdx.x * blockDim.x + threadIdx.x;
    if (n >= NTOK) return;
#pragma unroll
    for (int k = 0; k < 2; ++k) {
        int e = tok_idx[2 * n + k];
        float g = tok_gate[2 * n + k];
        int pos = atomicAdd(&cursor[e], 1);
        int slot = offsets[e] + pos;
        a_tok[slot] = n;
        a_gate[slot] = g;
    }
}

// ---------------- 6. expert FFN via fp32 WMMA ----------------
// 4 waves; wave wj owns ALL 32 rows (2 row-tiles) and a column slice, so every
// B fragment load feeds 2 wmma ops.
__launch_bounds__(128)
__global__ void ffn_kernel(const float* __restrict__ x,
                           const float* __restrict__ W1, const float* __restrict__ b1,
                           const float* __restrict__ W2, const float* __restrict__ b2,
                           const int* __restrict__ counts, const int* __restrict__ offsets,
                           const int* __restrict__ a_tok, const float* __restrict__ a_gate,
                           float* __restrict__ moe_out) {
    __shared__ int   tok_s[MT];
    __shared__ float gate_s[MT];
    __shared__ float Hc[MT * HCS];

    int e = blockIdx.x / TPE;
    int t = blockIdx.x % TPE;
    int cnt = counts[e];
    int m0 = t * MT;
    if (m0 >= cnt) return;           // uniform per block: WMMA below keeps EXEC all-1s

    int tid = threadIdx.x;
    if (tid < MT) {
        int gi = m0 + tid;
        bool valid = gi < cnt;
        tok_s[tid]  = valid ? a_tok[offsets[e] + gi] : 0;
        gate_s[tid] = valid ? a_gate[offsets[e] + gi] : 0.0f;
    }
    __syncthreads();

    int wj   = tid >> 5;             // wave id 0..3 = column slice
    int lane = tid & 31;
    int l15  = lane & 15;
    int hi16 = lane >> 4;            // 0 for lanes 0-15, 1 for 16-31
    int koff = hi16 << 1;            // K offset 0 / 2 for A,B fragments
    int roff = hi16 << 3;            // row offset 0 / 8 for C,D fragments

    // A-fragment row pointers for the two row tiles (rows l15 and 16+l15)
    const float* xrow0 = x + (size_t)tok_s[l15]      * DDIM + koff;
    const float* xrow1 = x + (size_t)tok_s[16 + l15] * DDIM + koff;
    const float* w1e  = W1 + (size_t)e * DDIM * HDIM;
    const float* w2e  = W2 + (size_t)e * HDIM * DDIM;
    const float* b1e  = b1 + (size_t)e * HDIM;
    const float* b2e  = b2 + (size_t)e * DDIM;

    v8f acc[16];                     // [ri][c]: ri in {0,1} row tile, c in 0..7 col tile
    const v8f vzero = {0.f, 0.f, 0.f, 0.f, 0.f, 0.f, 0.f, 0.f};
#pragma unroll
    for (int c = 0; c < 16; ++c) acc[c] = vzero;

    for (int hc0 = 0; hc0 < HDIM; hc0 += 128) {
        // ---- GEMM1: Hc[32][128] = gelu(X_tile @ W1e[:, hc0:hc0+128) + b1) ----
        // wave wj: hcols 32*wj .. 32*wj+31 (2 col tiles), all 32 rows (2 row tiles)
        v8f hacc[4];                 // [ri*2 + j]
#pragma unroll
        for (int j = 0; j < 4; ++j) hacc[j] = vzero;

        for (int k = 0; k < DDIM; k += 4) {
            v2f A0, A1;
            A0.x = xrow0[k];  A0.y = xrow0[k + 1];
            A1.x = xrow1[k];  A1.y = xrow1[k + 1];
            const float* wp = w1e + (size_t)(k + koff) * HDIM + hc0 + 32 * wj + l15;
#pragma unroll
            for (int j = 0; j < 2; ++j) {
                v2f B;
                B.x = wp[16 * j];           // W1e[k+koff  ][col]
                B.y = wp[16 * j + HDIM];    // W1e[k+koff+1][col]
                hacc[0 * 2 + j] = __builtin_amdgcn_wmma_f32_16x16x4_f32(
                    false, A0, false, B, (short)0, hacc[0 * 2 + j], false, false);
                hacc[1 * 2 + j] = __builtin_amdgcn_wmma_f32_16x16x4_f32(
                    false, A1, false, B, (short)0, hacc[1 * 2 + j], false, false);
            }
        }

        __syncthreads();   // prior GEMM2 reads of Hc done before overwrite
#pragma unroll
        for (int j = 0; j < 2; ++j) {
            int col = 32 * wj + 16 * j + l15;
            float bb = b1e[hc0 + col];
#pragma unroll
            for (int ri = 0; ri < 2; ++ri) {
#pragma unroll
                for (int r = 0; r < 8; ++r) {
                    int row = 16 * ri + r + roff;
                    Hc[row * HCS + col] = gelu_tanh(hacc[ri * 2 + j][r] + bb);
                }
            }
        }
        __syncthreads();

        // ---- GEMM2 partial: acc += Hc(32x128) @ W2e[hc0:hc0+128][:] ----
        // wave wj: dcols 128*wj .. 128*wj+127 (8 col tiles), all 32 rows
        for (int kk = 0; kk < 128; kk += 4) {
            v2f A0, A1;
            A0.x = Hc[l15 * HCS + kk + koff];
            A0.y = Hc[l15 * HCS + kk + koff + 1];
            A1.x = Hc[(16 + l15) * HCS + kk + koff];
            A1.y = Hc[(16 + l15) * HCS + kk + koff + 1];
            const float* wp = w2e + (size_t)(hc0 + kk + koff) * DDIM + 128 * wj + l15;
#pragma unroll
            for (int c = 0; c < 8; ++c) {
                v2f B;
                B.x = wp[16 * c];           // W2e[h  ][dcol]
                B.y = wp[16 * c + DDIM];    // W2e[h+1][dcol]
                acc[0 * 8 + c] = __builtin_amdgcn_wmma_f32_16x16x4_f32(
                    false, A0, false, B, (short)0, acc[0 * 8 + c], false, false);
                acc[1 * 8 + c] = __builtin_amdgcn_wmma_f32_16x16x4_f32(
                    false, A1, false, B, (short)0, acc[1 * 8 + c], false, false);
            }
        }
    }

    __syncthreads();
    // ---- epilogue: gate-scaled scatter-add (each moe_out element gets exactly 2 adds) ----
#pragma unroll
    for (int c = 0; c < 8; ++c) {
        int col = 128 * wj + 16 * c + l15;
        float bb = b2e[col];
#pragma unroll
        for (int ri = 0; ri < 2; ++ri) {
#pragma unroll
            for (int r = 0; r < 8; ++r) {
                int m = 16 * ri + r + roff;
                float g = gate_s[m];
                if (g != 0.0f) {
                    atomicAdd(&moe_out[(size_t)tok_s[m] * DDIM + col],
                              g * (acc[ri * 8 + c][r] + bb));
                }
            }
        }
    }
}

// ---------------- 7. residual + L2 norm + gelu (128-bit accesses) ----------------
__global__ void finalize_kernel(const float* __restrict__ x, const float* __restrict__ moe,
                                const float* __restrict__ gamma, float* __restrict__ out) {
    int n = blockIdx.x;
    int tid = threadIdx.x;     // 128 threads, 4 consecutive elements each
    int lane = tid & 31;
    int wave = tid >> 5;

    const float4* x4 = (const float4*)(x   + (size_t)n * DDIM);
    const float4* m4 = (const float4*)(moe + (size_t)n * DDIM);
    const float4* g4 = (const float4*)gamma;

    float4 xv = x4[tid];
    float4 mv = m4[tid];
    float y0 = xv.x + mv.x;
    float y1 = xv.y + mv.y;
    float y2 = xv.z + mv.z;
    float y3 = xv.w + mv.w;
    float ss = y0 * y0 + y1 * y1 + y2 * y2 + y3 * y3;

#pragma unroll
    for (int off = 16; off > 0; off >>= 1) ss += __shfl_xor(ss, off, 32);

    __shared__ float wsum[4];
    if (lane == 0) wsum[wave] = ss;
    __syncthreads();
    float tot = wsum[0] + wsum[1] + wsum[2] + wsum[3];
    float norm = sqrtf(tot);
    float scale = sqrtf((float)DDIM) / fmaxf(norm, 1e-12f);

    float4 gv = g4[tid];
    float4 o;
    o.x = gelu_tanh(y0 * scale * gv.x);
    o.y = gelu_tanh(y1 * scale * gv.y);
    o.z = gelu_tanh(y2 * scale * gv.z);
    o.w = gelu_tanh(y3 * scale * gv.w);
    ((float4*)(out + (size_t)n * DDIM))[tid] = o;
}

// ---------------- host launcher ----------------
extern "C" void kernel_launch(void* const* d_in, const int* in_sizes, int n_in,
                              void* d_out, int out_size, void* d_ws, size_t ws_size,
                              hipStream_t stream) {
    const float* x     = (const float*)d_in[0];
    const float* wgate = (const float*)d_in[1];
    const float* W1    = (const float*)d_in[2];
    const float* b1    = (const float*)d_in[3];
    const float* W2    = (const float*)d_in[4];
    const float* b2    = (const float*)d_in[5];
    const float* gamma = (const float*)d_in[6];
    float* out = (float*)d_out;

    // workspace layout
    char* ws = (char*)d_ws;
    float* moe      = (float*)ws;                               // N*D floats (16 MB)
    size_t off = (size_t)NTOK * DDIM * sizeof(float);
    int*   tok_idx  = (int*)(ws + off);   off += 2 * NTOK * sizeof(int);
    float* tok_gate = (float*)(ws + off); off += 2 * NTOK * sizeof(float);
    int*   a_tok    = (int*)(ws + off);   off += 2 * NTOK * sizeof(int);
    float* a_gate   = (float*)(ws + off); off += 2 * NTOK * sizeof(float);
    int*   counts   = (int*)(ws + off);   off += EEXP * sizeof(int);
    int*   offsets  = (int*)(ws + off);   off += EEXP * sizeof(int);
    int*   cursor   = (int*)(ws + off);   off += EEXP * sizeof(int);
    float* imp      = (float*)(ws + off); off += EEXP * sizeof(float);
    float* loadv    = (float*)(ws + off); off += EEXP * sizeof(float);
    (void)ws_size; (void)in_sizes; (void)n_in; (void)out_size;

    zero_kernel<<<1024, 256, 0, stream>>>((float4*)moe, (size_t)NTOK * DDIM / 4);
    gate_kernel<<<NTOK / 8, 256, 0, stream>>>(x, wgate, tok_idx, tok_gate);
    stats_kernel<<<EEXP, 256, 0, stream>>>(tok_idx, tok_gate, counts, imp, loadv);
    scan_loss_kernel<<<1, 32, 0, stream>>>(counts, offsets, cursor, imp, loadv,
                                           out + (size_t)NTOK * DDIM);
    scatter_kernel<<<NTOK / 256, 256, 0, stream>>>(tok_idx, tok_gate, offsets, cursor,
                                                   a_tok, a_gate);
    ffn_kernel<<<EEXP * TPE, 128, 0, stream>>>(x, W1, b1, W2, b2, counts, offsets,
                                               a_tok, a_gate, moe);
    finalize_kernel<<<NTOK, 128, 0, stream>>>(x, moe, gamma, out);
}